// Network_515396076038
// MI455X (gfx1250) — compile-verified
//
#include <hip/hip_runtime.h>
#include <cstddef>

// ---------------------------------------------------------------------------
// MI455X (gfx1250) implementation.
//
// Heavy work: 5 tall-skinny f32 GEMMs, N=4096 rows, K=4096, C in {20,21}.
// ~3.4 GFLOP vs ~0.2 GB mandatory HBM traffic -> ~400 TFLOP/s needed to be
// bandwidth bound; V_WMMA_F32_16X16X4_F32 clears that, so full f32 precision.
//
// GEMM kernel: 4 waves/block, each wave owns a 32x32 output tile (4 WMMA
// accumulators = 4 independent dependency chains). K staged in 32-wide chunks
// through double-buffered LDS with a software pipeline; staging in two
// register halves (4xfloat4 + 4xfloat) to bound VGPR pressure. All buffer
// indices are literals (macro-expanded bodies) so every LDS address is a
// per-thread invariant + compile-time immediate. W staging loads are
// unconditional (clamped column) + cndmask-zeroed to avoid divergent
// saveexec branches. LDS stride 36: 36*r mod 64 tiles the banks in exact
// 4-bank windows -> conflict-free b64 fragment loads, 16B-aligned b128
// staging stores. B is stored transposed so each lane's fragment is one
// contiguous float2 (single ds_load_b64 into an even-aligned VGPR pair).
// ---------------------------------------------------------------------------

typedef float v2f __attribute__((ext_vector_type(2)));
typedef float v8f __attribute__((ext_vector_type(8)));

#define NROWS 4096
#define KDIM  4096
#define LDSTR 36   // LDS row stride (floats): conflict-free + 16B aligned

// ---------------------------------------------------------------------------
// Kernel 1: five GEMMs. grid = (32 row-tiles, 5 gemm-ids), block = 128 (4 waves)
// ---------------------------------------------------------------------------
__global__ __launch_bounds__(128, 1) void gemm5_kernel(
    const float* __restrict__ roi, const float* __restrict__ ctxf,
    const float* __restrict__ frm,
    const float* __restrict__ Wcls, const float* __restrict__ bcls,
    const float* __restrict__ Wdet,
    const float* __restrict__ Wr1,  const float* __restrict__ br1,
    const float* __restrict__ Wr2,  const float* __restrict__ br2,
    float* __restrict__ scls, float* __restrict__ sfr, float* __restrict__ sctx,
    float* __restrict__ sr1,  float* __restrict__ sr2)
{
    __shared__ __align__(16) float lA[2][128 * LDSTR];  // rows x K, 2 buffers
    __shared__ __align__(16) float lB[2][32 * LDSTR];   // cols x K (transposed)

    const float* Aptr; const float* Wptr; const float* bptr; float* Optr; int C;
    switch (blockIdx.y) {
        case 0:  Aptr = roi;  Wptr = Wcls; bptr = bcls;    Optr = scls; C = 20; break;
        case 1:  Aptr = frm;  Wptr = Wdet; bptr = nullptr; Optr = sfr;  C = 20; break;
        case 2:  Aptr = ctxf; Wptr = Wdet; bptr = nullptr; Optr = sctx; C = 20; break;
        case 3:  Aptr = roi;  Wptr = Wr1;  bptr = br1;     Optr = sr1;  C = 21; break;
        default: Aptr = roi;  Wptr = Wr2;  bptr = br2;     Optr = sr2;  C = 21; break;
    }

    const int tid  = threadIdx.x;
    const int lane = tid & 31;
    const int wave = tid >> 5;
    const int lr   = lane & 15;   // 0..15
    const int lg   = lane >> 4;   // 0 or 1
    const int rowBase = blockIdx.x * 128;
    const int wrow = wave * 32;   // wave's 32 rows within the block tile

    // ---- invariant staging addresses ----
    // A: float4 element (row = tid>>3 + 16*i, col = (tid&7)*4 + k0)
    const int arow = tid >> 3;
    const int acp  = (tid & 7) * 4;
    const float* aSrc = Aptr + (size_t)(rowBase + arow) * KDIM + acp;  // += 32 per chunk
    // W: element (kk = tid>>5 + 4*i, cc = tid&31); clamped column -> always-safe
    // unconditional load, value zeroed by cndmask for the padding columns.
    const int wkk  = tid >> 5;
    const int wcc  = tid & 31;
    const bool wVal = (wcc < C);
    const int wccs = wVal ? wcc : (C - 1);
    const float* wSrc = Wptr + (size_t)wkk * C + wccs;                 // += 32*C per chunk

    const int aDst = arow * LDSTR + acp;       // element offset within lA[buf]
    const int bDst = wcc * LDSTR + wkk;        // element offset within lB[buf]

    // ---- invariant fragment addresses ----
    const int aOff = (wrow + lr) * LDSTR + 2 * lg;
    const int bOff = lr * LDSTR + 2 * lg;

    v8f acc00 = {}, acc01 = {}, acc10 = {}, acc11 = {};

    float4 ra[4];   // half of the A staging (rows tid>>3 + 16*{0..3} or {4..7})
    float  rb[4];   // half of the W staging

#define LOAD_HALF(H)                                                          \
    {                                                                         \
        _Pragma("unroll")                                                     \
        for (int j = 0; j < 4; ++j)                                           \
            ra[j] = *(const float4*)(aSrc + (size_t)(4*(H)+j) * 16 * KDIM);   \
        _Pragma("unroll")                                                     \
        for (int j = 0; j < 4; ++j) {                                         \
            float v = wSrc[(size_t)(4*(H)+j) * 4 * C];                        \
            rb[j] = wVal ? v : 0.0f;                                          \
        }                                                                     \
    }

#define STORE_HALF(NXT, H)                                                    \
    {                                                                         \
        _Pragma("unroll")                                                     \
        for (int j = 0; j < 4; ++j)                                           \
            *(float4*)(&lA[NXT][aDst] + (4*(H)+j) * 16 * LDSTR) = ra[j];      \
        _Pragma("unroll")                                                     \
        for (int j = 0; j < 4; ++j)                                           \
            lB[NXT][bDst + (4*(H)+j) * 4] = rb[j];                            \
    }

#define COMPUTE_HALF(CUR, KB)                                                 \
    {                                                                         \
        _Pragma("unroll")                                                     \
        for (int kk = (KB); kk < (KB) + 16; kk += 4) {                        \
            /* 16x16x4 f32 fragment: lanes 0-15 hold K={kk,kk+1},           */\
            /* lanes 16-31 hold K={kk+2,kk+3}; contiguous float2 per lane.  */\
            v2f a0 = *(const v2f*)(&lA[CUR][aOff] + kk);                      \
            v2f a1 = *(const v2f*)(&lA[CUR][aOff] + 16 * LDSTR + kk);         \
            v2f b0 = *(const v2f*)(&lB[CUR][bOff] + kk);                      \
            v2f b1 = *(const v2f*)(&lB[CUR][bOff] + 16 * LDSTR + kk);         \
            acc00 = __builtin_amdgcn_wmma_f32_16x16x4_f32(                    \
                false, a0, false, b0, (short)0, acc00, false, false);         \
            acc01 = __builtin_amdgcn_wmma_f32_16x16x4_f32(                    \
                false, a0, false, b1, (short)0, acc01, false, false);         \
            acc10 = __builtin_amdgcn_wmma_f32_16x16x4_f32(                    \
                false, a1, false, b0, (short)0, acc10, false, false);         \
            acc11 = __builtin_amdgcn_wmma_f32_16x16x4_f32(                    \
                false, a1, false, b1, (short)0, acc11, false, false);         \
        }                                                                     \
    }

#define ADVANCE_SRC() { aSrc += 32; wSrc += (size_t)32 * C; }

    // prologue: stage chunk 0 into buffer 0
    LOAD_HALF(0); STORE_HALF(0, 0);
    LOAD_HALF(1); STORE_HALF(0, 1);
    ADVANCE_SRC();
    __syncthreads();

    const int nch = KDIM / 32;   // 128 (even)
    for (int c = 0; c < nch; c += 2) {
        // chunk c from buffer 0; stage chunk c+1 (always exists) into buffer 1
        LOAD_HALF(0);
        COMPUTE_HALF(0, 0);
        STORE_HALF(1, 0);
        LOAD_HALF(1);
        COMPUTE_HALF(0, 16);
        STORE_HALF(1, 1);
        ADVANCE_SRC();
        __syncthreads();

        // chunk c+1 from buffer 1; stage chunk c+2 into buffer 0 if it exists
        if (c + 2 < nch) {
            LOAD_HALF(0);
            COMPUTE_HALF(1, 0);
            STORE_HALF(0, 0);
            LOAD_HALF(1);
            COMPUTE_HALF(1, 16);
            STORE_HALF(0, 1);
            ADVANCE_SRC();
            __syncthreads();
        } else {
            COMPUTE_HALF(1, 0);
            COMPUTE_HALF(1, 16);
            __syncthreads();
        }
    }

#undef LOAD_HALF
#undef STORE_HALF
#undef COMPUTE_HALF
#undef ADVANCE_SRC

    // epilogue: C/D layout -> VGPR i holds M=i (lanes 0-15) / M=i+8 (lanes 16-31)
    #pragma unroll
    for (int i = 0; i < 8; ++i) {
        int m  = i + 8 * lg;
        int r0 = rowBase + wrow + m;
        int r1 = r0 + 16;
        if (lr < C) {
            float b = bptr ? bptr[lr] : 0.0f;
            Optr[(size_t)r0 * C + lr] = acc00[i] + b;
            Optr[(size_t)r1 * C + lr] = acc10[i] + b;
        }
        if (lr + 16 < C) {
            float b = bptr ? bptr[lr + 16] : 0.0f;
            Optr[(size_t)r0 * C + lr + 16] = acc01[i] + b;
            Optr[(size_t)r1 * C + lr + 16] = acc11[i] + b;
        }
    }
}

// ---------------------------------------------------------------------------
// Kernel 2: det = frame - context ; column-softmax stats (max, sum-exp) per col
// ---------------------------------------------------------------------------
__global__ __launch_bounds__(256) void det_colstats_kernel(
    const float* __restrict__ sfr, const float* __restrict__ sctx,
    float* __restrict__ det, float* __restrict__ colmax, float* __restrict__ colsum)
{
    __shared__ float red[256];
    const int c = blockIdx.x;       // 0..19
    const int tid = threadIdx.x;

    float m = -3.402823e38f;
    for (int n = tid; n < NROWS; n += 256) {
        float d = sfr[n * 20 + c] - sctx[n * 20 + c];
        det[n * 20 + c] = d;
        m = fmaxf(m, d);
    }
    red[tid] = m; __syncthreads();
    for (int s = 128; s > 0; s >>= 1) {
        if (tid < s) red[tid] = fmaxf(red[tid], red[tid + s]);
        __syncthreads();
    }
    float cm = red[0]; __syncthreads();

    float sum = 0.0f;
    for (int n = tid; n < NROWS; n += 256) sum += expf(det[n * 20 + c] - cm);
    red[tid] = sum; __syncthreads();
    for (int s = 128; s > 0; s >>= 1) {
        if (tid < s) red[tid] += red[tid + s];
        __syncthreads();
    }
    if (tid == 0) { colmax[c] = cm; colsum[c] = red[0]; }
}

// ---------------------------------------------------------------------------
// Kernel 3: per-row softmaxes + derived products. One thread per row.
// ---------------------------------------------------------------------------
__global__ __launch_bounds__(256) void rowwise_kernel(
    const float* __restrict__ scls, const float* __restrict__ sr1,
    const float* __restrict__ sr2,  const float* __restrict__ det,
    const float* __restrict__ colmax, const float* __restrict__ colsum,
    const float* __restrict__ isw,
    float* __restrict__ dprob, float* __restrict__ p1, float* __restrict__ p2,
    float* __restrict__ rp1, float* __restrict__ rp2)
{
    int n = blockIdx.x * blockDim.x + threadIdx.x;
    if (n >= NROWS) return;
    const float w = isw[n];
    float e[21];

    // cls row softmax (20)
    float m = -3.402823e38f;
    #pragma unroll
    for (int c = 0; c < 20; ++c) m = fmaxf(m, scls[n * 20 + c]);
    float s = 0.0f;
    #pragma unroll
    for (int c = 0; c < 20; ++c) { e[c] = expf(scls[n * 20 + c] - m); s += e[c]; }
    float inv = 1.0f / s;

    // det_prob (column softmax) and p1 = cls_prob * det_prob * ISw
    #pragma unroll
    for (int c = 0; c < 20; ++c) {
        float dp = expf(det[n * 20 + c] - colmax[c]) / colsum[c];
        dprob[n * 20 + c] = dp;
        p1[n * 20 + c] = (e[c] * inv) * dp * w;
    }

    // refine_prob_1 row softmax (21) ; p2 = rp1[:,1:] * ISw
    m = -3.402823e38f;
    #pragma unroll
    for (int c = 0; c < 21; ++c) m = fmaxf(m, sr1[n * 21 + c]);
    s = 0.0f;
    #pragma unroll
    for (int c = 0; c < 21; ++c) { e[c] = expf(sr1[n * 21 + c] - m); s += e[c]; }
    inv = 1.0f / s;
    #pragma unroll
    for (int c = 0; c < 21; ++c) rp1[n * 21 + c] = e[c] * inv;
    #pragma unroll
    for (int c = 0; c < 20; ++c) p2[n * 20 + c] = rp1[n * 21 + c + 1] * w;

    // refine_prob_2 row softmax (21)
    m = -3.402823e38f;
    #pragma unroll
    for (int c = 0; c < 21; ++c) m = fmaxf(m, sr2[n * 21 + c]);
    s = 0.0f;
    #pragma unroll
    for (int c = 0; c < 21; ++c) { e[c] = expf(sr2[n * 21 + c] - m); s += e[c]; }
    inv = 1.0f / s;
    #pragma unroll
    for (int c = 0; c < 21; ++c) rp2[n * 21 + c] = e[c] * inv;
}

// ---------------------------------------------------------------------------
// Kernel 4: per column c: argmax_n p1[:,c], argmax_n p2[:,c] (first-index ties,
// matching jnp.argmax), det_cls_score[c] = sum_n cls_score*det_prob, and
// extraction of the query boxes from ss_boxes.
// ---------------------------------------------------------------------------
__global__ __launch_bounds__(256) void colstats2_kernel(
    const float* __restrict__ p1, const float* __restrict__ p2,
    const float* __restrict__ scls, const float* __restrict__ dprob,
    const float* __restrict__ ssb,
    float* __restrict__ dcs, float* __restrict__ boxes1, float* __restrict__ boxes2,
    int* __restrict__ midx1, int* __restrict__ midx2)
{
    __shared__ float sv[256];
    __shared__ int   si[256];
    const int c = blockIdx.x;
    const int tid = threadIdx.x;

    // ---- argmax of p1[:,c] ----
    float bv = -3.402823e38f; int bi = 0x7fffffff;
    for (int n = tid; n < NROWS; n += 256) {
        float v = p1[n * 20 + c];
        if (v > bv) { bv = v; bi = n; }
    }
    sv[tid] = bv; si[tid] = bi; __syncthreads();
    for (int s = 128; s > 0; s >>= 1) {
        if (tid < s) {
            float v2 = sv[tid + s]; int i2 = si[tid + s];
            if (v2 > sv[tid] || (v2 == sv[tid] && i2 < si[tid])) { sv[tid] = v2; si[tid] = i2; }
        }
        __syncthreads();
    }
    if (tid == 0) {
        int idx = si[0];
        midx1[c] = idx;
        #pragma unroll
        for (int j = 0; j < 4; ++j) boxes1[c * 4 + j] = ssb[idx * 5 + 1 + j];
    }
    __syncthreads();

    // ---- argmax of p2[:,c] ----
    bv = -3.402823e38f; bi = 0x7fffffff;
    for (int n = tid; n < NROWS; n += 256) {
        float v = p2[n * 20 + c];
        if (v > bv) { bv = v; bi = n; }
    }
    sv[tid] = bv; si[tid] = bi; __syncthreads();
    for (int s = 128; s > 0; s >>= 1) {
        if (tid < s) {
            float v2 = sv[tid + s]; int i2 = si[tid + s];
            if (v2 > sv[tid] || (v2 == sv[tid] && i2 < si[tid])) { sv[tid] = v2; si[tid] = i2; }
        }
        __syncthreads();
    }
    if (tid == 0) {
        int idx = si[0];
        midx2[c] = idx;
        #pragma unroll
        for (int j = 0; j < 4; ++j) boxes2[c * 4 + j] = ssb[idx * 5 + 1 + j];
    }
    __syncthreads();

    // ---- det_cls_score[c] ----
    float sum = 0.0f;
    for (int n = tid; n < NROWS; n += 256)
        sum += scls[n * 20 + c] * dprob[n * 20 + c];
    sv[tid] = sum; __syncthreads();
    for (int s = 128; s > 0; s >>= 1) {
        if (tid < s) sv[tid] += sv[tid + s];
        __syncthreads();
    }
    if (tid == 0) dcs[c] = sv[0];
}

// ---------------------------------------------------------------------------
// Kernel 5: per-row refine supervision + loss accumulation.
// accum = [num1, cnt1, num2, cnt2]
// ---------------------------------------------------------------------------
__device__ __forceinline__ void refine_row(
    float bx1, float by1, float bx2, float by2, float area_b,
    const float* __restrict__ qboxes, const int* __restrict__ label,
    const float* __restrict__ rp, int n, float w, float* num, float* cnt)
{
    float mo = -1.0f; int gt = 0;
    #pragma unroll
    for (int c = 0; c < 20; ++c) {
        float ov = -1.0f;
        if (label[c] == 1) {
            float qx1 = qboxes[c * 4 + 0], qy1 = qboxes[c * 4 + 1];
            float qx2 = qboxes[c * 4 + 2], qy2 = qboxes[c * 4 + 3];
            float iw = fminf(bx2, qx2) - fmaxf(bx1, qx1) + 1.0f;
            float ih = fminf(by2, qy2) - fmaxf(by1, qy1) + 1.0f;
            iw = fmaxf(iw, 0.0f); ih = fmaxf(ih, 0.0f);
            float inter = iw * ih;
            float area_q = (qx2 - qx1 + 1.0f) * (qy2 - qy1 + 1.0f);
            ov = inter / (area_b + area_q - inter);
        }
        if (ov > mo) { mo = ov; gt = c; }
    }
    bool fg = mo > 0.5f;
    bool bg = (mo >= 0.1f) && (mo < 0.5f);
    if (fg || bg) {
        int col = fg ? gt + 1 : 0;
        *num = w * logf(rp[n * 21 + col]);
        *cnt = 1.0f;
    }
}

__global__ __launch_bounds__(256) void loss_rows_kernel(
    const float* __restrict__ ssb, const int* __restrict__ label,
    const float* __restrict__ isw,
    const float* __restrict__ boxes1, const float* __restrict__ boxes2,
    const float* __restrict__ rp1, const float* __restrict__ rp2,
    float* __restrict__ accum)
{
    __shared__ float s0[256], s1[256], s2[256], s3[256];
    const int tid = threadIdx.x;
    const int n = blockIdx.x * 256 + tid;
    float num1 = 0.0f, c1 = 0.0f, num2 = 0.0f, c2 = 0.0f;
    if (n < NROWS) {
        float bx1 = ssb[n * 5 + 1], by1 = ssb[n * 5 + 2];
        float bx2 = ssb[n * 5 + 3], by2 = ssb[n * 5 + 4];
        float area_b = (bx2 - bx1 + 1.0f) * (by2 - by1 + 1.0f);
        float w = isw[n];
        refine_row(bx1, by1, bx2, by2, area_b, boxes1, label, rp1, n, w, &num1, &c1);
        refine_row(bx1, by1, bx2, by2, area_b, boxes2, label, rp2, n, w, &num2, &c2);
    }
    s0[tid] = num1; s1[tid] = c1; s2[tid] = num2; s3[tid] = c2;
    __syncthreads();
    for (int s = 128; s > 0; s >>= 1) {
        if (tid < s) {
            s0[tid] += s0[tid + s]; s1[tid] += s1[tid + s];
            s2[tid] += s2[tid + s]; s3[tid] += s3[tid + s];
        }
        __syncthreads();
    }
    if (tid == 0) {
        atomicAdd(&accum[0], s0[0]); atomicAdd(&accum[1], s1[0]);
        atomicAdd(&accum[2], s2[0]); atomicAdd(&accum[3], s3[0]);
    }
}

// ---------------------------------------------------------------------------
// Kernel 6: final scalar loss
// ---------------------------------------------------------------------------
__global__ void finalize_kernel(const float* __restrict__ dcs,
                                const int* __restrict__ label,
                                const float* __restrict__ accum,
                                float* __restrict__ out)
{
    if (threadIdx.x == 0 && blockIdx.x == 0) {
        float loss = 0.0f;
        for (int c = 0; c < 20; ++c) {
            float lab = (float)label[c];
            loss += fmaxf(0.0f, 1.0f - lab * dcs[c]);
        }
        loss *= (1.0f / 20.0f);
        float rl1 = -accum[0] / accum[1];
        float rl2 = -accum[2] / accum[3];
        out[0] = loss + 0.1f * rl1 + 0.1f * rl2;
    }
}

// ---------------------------------------------------------------------------
// Launch
// ---------------------------------------------------------------------------
extern "C" void kernel_launch(void* const* d_in, const int* in_sizes, int n_in,
                              void* d_out, int out_size, void* d_ws, size_t ws_size,
                              hipStream_t stream) {
    (void)in_sizes; (void)n_in; (void)out_size; (void)ws_size;

    const float* roi  = (const float*)d_in[0];
    const float* fctx = (const float*)d_in[1];
    const float* ffrm = (const float*)d_in[2];
    const float* Wcls = (const float*)d_in[3];
    const float* bcls = (const float*)d_in[4];
    const float* Wdet = (const float*)d_in[5];
    /* b_det (d_in[6]) cancels in frame - context */
    const float* Wr1  = (const float*)d_in[7];
    const float* br1  = (const float*)d_in[8];
    const float* Wr2  = (const float*)d_in[9];
    const float* br2  = (const float*)d_in[10];
    const float* ssb  = (const float*)d_in[11];
    const float* isw  = (const float*)d_in[12];
    const int*   lbl  = (const int*)d_in[13];

    float* ws = (float*)d_ws;
    const size_t N20 = (size_t)NROWS * 20;
    const size_t N21 = (size_t)NROWS * 21;
    size_t off = 0;
    float* scls   = ws + off; off += N20;
    float* sfr    = ws + off; off += N20;
    float* sctx   = ws + off; off += N20;
    float* sr1    = ws + off; off += N21;
    float* sr2    = ws + off; off += N21;
    float* rp1    = ws + off; off += N21;
    float* rp2    = ws + off; off += N21;
    float* det    = ws + off; off += N20;
    float* dprob  = ws + off; off += N20;
    float* p1     = ws + off; off += N20;
    float* p2     = ws + off; off += N20;
    float* colmax = ws + off; off += 32;
    float* colsum = ws + off; off += 32;
    float* dcs    = ws + off; off += 32;
    float* boxes1 = ws + off; off += 96;
    float* boxes2 = ws + off; off += 96;
    float* accum  = ws + off; off += 32;
    int*   midx1  = (int*)(ws + off); off += 32;
    int*   midx2  = (int*)(ws + off); off += 32;

    hipMemsetAsync(accum, 0, 4 * sizeof(float), stream);

    gemm5_kernel<<<dim3(32, 5), 128, 0, stream>>>(
        roi, fctx, ffrm, Wcls, bcls, Wdet, Wr1, br1, Wr2, br2,
        scls, sfr, sctx, sr1, sr2);

    det_colstats_kernel<<<20, 256, 0, stream>>>(sfr, sctx, det, colmax, colsum);

    rowwise_kernel<<<16, 256, 0, stream>>>(
        scls, sr1, sr2, det, colmax, colsum, isw, dprob, p1, p2, rp1, rp2);

    colstats2_kernel<<<20, 256, 0, stream>>>(
        p1, p2, scls, dprob, ssb, dcs, boxes1, boxes2, midx1, midx2);

    loss_rows_kernel<<<16, 256, 0, stream>>>(
        ssb, lbl, isw, boxes1, boxes2, rp1, rp2, accum);

    finalize_kernel<<<1, 32, 0, stream>>>(dcs, lbl, accum, (float*)d_out);
}